// TransferNet_59261958750264
// MI455X (gfx1250) — compile-verified
//
#include <hip/hip_runtime.h>
#include <hip/hip_bf16.h>

// ---------------- problem constants ----------------
#define BSZ 8
#define NE  40000
#define DD  128
#define RR  512
#define KK  3
#define DEG 64
#define MM  (KK * DEG)        // 192 gathered triples per example per step
#define NROWS (BSZ * MM)      // 1536 GRU rows per step
#define G3D (3 * DD)          // 384 gate outputs

typedef __attribute__((ext_vector_type(2))) float v2f;
typedef __attribute__((ext_vector_type(8))) float v8f;

__device__ __forceinline__ float sigf(float x) { return 1.0f / (1.0f + expf(-x)); }
__device__ __forceinline__ float clamp1(float v) { return v > 1.0f ? 1.0f : v; }

// ---------------- zero ----------------
__global__ void zero_kernel(float* p, int n) {
    int i = blockIdx.x * blockDim.x + threadIdx.x;
    if (i < n) p[i] = 0.0f;
}

// ---------------- cq[t][b][:] = tanh(rel_emb[query[b]] @ step_W[t] + step_b[t]) ----------------
__global__ void cq_kernel(const float* __restrict__ rel_emb, const int* __restrict__ query,
                          const float* __restrict__ step_W, const float* __restrict__ step_b,
                          float* __restrict__ cq) {
    int tb = blockIdx.x;           // t*8 + b, 16 blocks
    int t = tb >> 3, b = tb & 7;
    int d = threadIdx.x;           // 128 threads
    const float* q = rel_emb + (size_t)query[b] * DD;
    const float* W = step_W + (size_t)t * DD * DD;
    float acc = step_b[t * DD + d];
    #pragma unroll 4
    for (int i = 0; i < DD; ++i) acc += q[i] * W[i * DD + d];
    cq[tb * DD + d] = tanhf(acc);
}

// ---------------- top-3 per example, value desc / index asc (matches jax.lax.top_k) ----------------
__global__ void topk3_kernel(const float* __restrict__ vals, int* __restrict__ topk) {
    int b = blockIdx.x;            // 8 blocks
    int tid = threadIdx.x;         // 256 threads
    __shared__ float sv[256];
    __shared__ int   si[256];
    __shared__ int   chosen[KK];
    for (int p = 0; p < KK; ++p) {
        float bestv = -1.0f; int besti = NE;
        for (int i = tid; i < NE; i += 256) {
            if (p > 0 && i == chosen[0]) continue;
            if (p > 1 && i == chosen[1]) continue;
            float v = clamp1(vals[(size_t)b * NE + i]);   // normalization applied on read
            if (v > bestv || (v == bestv && i < besti)) { bestv = v; besti = i; }
        }
        sv[tid] = bestv; si[tid] = besti;
        __syncthreads();
        for (int s = 128; s > 0; s >>= 1) {
            if (tid < s) {
                if (sv[tid + s] > sv[tid] || (sv[tid + s] == sv[tid] && si[tid + s] < si[tid])) {
                    sv[tid] = sv[tid + s]; si[tid] = si[tid + s];
                }
            }
            __syncthreads();
        }
        if (tid == 0) { chosen[p] = si[0]; topk[b * KK + p] = si[0]; }
        __syncthreads();
    }
}

// ---------------- gather triples; build X (rel_emb rows), H (sparse hist rows), sub_p, obj ----------------
__global__ void gather_kernel(const int* __restrict__ topk, int nvalid,
                              const float* __restrict__ prev_e,       // [BSZ][NE] raw
                              const float* __restrict__ rel_emb,
                              const int* __restrict__ kb_triple, const int* __restrict__ kb_range,
                              const float* __restrict__ hbk,          // [BSZ*KK][DD] or nullptr
                              float* __restrict__ X, float* __restrict__ H,
                              float* __restrict__ sub_p, int* __restrict__ obj_idx) {
    int r = blockIdx.x;            // 1536 blocks
    int d = threadIdx.x;           // 128 threads
    int b = r / MM, m = r % MM, k = m / DEG, j = m % DEG;
    int e = topk[b * KK + k];
    int lo = kb_range[2 * e], hi = kb_range[2 * e + 1];
    int tri = lo + j;
    bool valid = (k < nvalid) && (tri < hi);
    if (!valid) tri = 0;                       // mirrors jnp.where(valid, tri_idx, 0)
    int obj = kb_triple[(size_t)tri * 3 + 1];
    int rel = kb_triple[(size_t)tri * 3 + 2];
    X[(size_t)r * DD + d] = rel_emb[(size_t)rel * DD + d];
    H[(size_t)r * DD + d] = hbk ? hbk[(b * KK + k) * DD + d] : 0.0f;
    if (d == 0) {
        float pe = clamp1(prev_e[(size_t)b * NE + e]);   // normalized last_e
        sub_p[r] = valid ? pe : 0.0f;                    // fold valid mask into sub_p
        obj_idx[r] = obj;
    }
}

// ---------------- WMMA f32 GEMM: GI = X @ W_ih^T, GH = H @ W_hh^T (biases added later) ----------------
// One wave per 16x16 output tile. V_WMMA_F32_16X16X4_F32, K-loop of 32 over D=128.
// A 16x4 layout (ISA 7.12.2): lanes 0-15 hold K0/K1, lanes 16-31 hold K2/K3; B mirrors with N across lanes.
__global__ void __launch_bounds__(128) wmma_gemm_kernel(
        const float* __restrict__ X, const float* __restrict__ H,
        const float* __restrict__ Wih, const float* __restrict__ Whh,
        float* __restrict__ GI, float* __restrict__ GH) {
    const int TR = NROWS / 16;     // 96
    const int TC = G3D / 16;       // 24
    int wave = blockIdx.x * (blockDim.x >> 5) + (threadIdx.x >> 5);
    int which = wave / (TR * TC);                 // 0: GI, 1: GH (uniform per wave)
    int rem = wave % (TR * TC);
    int tr = rem / TC, tc = rem % TC;
    const float* A = which ? H : X;
    const float* W = which ? Whh : Wih;           // W[o][d], row-major [384][128]
    float* Dst = which ? GH : GI;
    int lane = threadIdx.x & 31;
    int half = lane >> 4, l16 = lane & 15;
    const float* arow = A + (size_t)(tr * 16 + l16) * DD;
    const float* wrow = W + (size_t)(tc * 16 + l16) * DD; // B[k][n] = W[n][k]
    __builtin_prefetch(wrow, 0, 3);
    __builtin_prefetch(arow, 0, 1);
    v8f c = {0.f, 0.f, 0.f, 0.f, 0.f, 0.f, 0.f, 0.f};
    #pragma unroll 8
    for (int kb = 0; kb < DD; kb += 4) {
        v2f a, b;
        a.x = arow[kb + 2 * half];
        a.y = arow[kb + 2 * half + 1];
        b.x = wrow[kb + 2 * half];
        b.y = wrow[kb + 2 * half + 1];
        c = __builtin_amdgcn_wmma_f32_16x16x4_f32(false, a, false, b, (short)0, c, false, false);
    }
    // C/D layout: lane n = l16, row m = vgpr + 8*half
    float* drow = Dst + (size_t)(tr * 16 + 8 * half) * G3D + tc * 16 + l16;
    #pragma unroll
    for (int v = 0; v < 8; ++v) drow[(size_t)v * G3D] = c[v];
}

// ---------------- GRU gates, transition prob, scatter-add into last_e ----------------
__global__ void gru_gate_kernel(const float* __restrict__ GI, const float* __restrict__ GH,
                                const float* __restrict__ H,
                                const float* __restrict__ b_ih, const float* __restrict__ b_hh,
                                const float* __restrict__ cq,   // [BSZ][DD] for this step
                                const float* __restrict__ cls_w, const float* __restrict__ cls_b,
                                const float* __restrict__ sub_p, const int* __restrict__ obj_idx,
                                float* __restrict__ lastE_out,
                                float* __restrict__ trans_out, float* __restrict__ objp_out) {
    int r = blockIdx.x;            // 1536 blocks
    int d = threadIdx.x;           // 128 threads
    int b = r / MM;
    const float* gi = GI + (size_t)r * G3D;
    const float* gh = GH + (size_t)r * G3D;
    float i_r = gi[d]            + b_ih[d];
    float i_z = gi[DD + d]       + b_ih[DD + d];
    float i_n = gi[2 * DD + d]   + b_ih[2 * DD + d];
    float h_r = gh[d]            + b_hh[d];
    float h_z = gh[DD + d]       + b_hh[DD + d];
    float h_n = gh[2 * DD + d]   + b_hh[2 * DD + d];
    float rg = sigf(i_r + h_r);
    float zg = sigf(i_z + h_z);
    float ng = tanhf(i_n + rg * h_n);
    float h  = H[(size_t)r * DD + d];
    float tr = (1.0f - zg) * ng + zg * h;
    __shared__ float red[DD];
    red[d] = tr * cq[b * DD + d] * cls_w[d];
    __syncthreads();
    for (int s = 64; s > 0; s >>= 1) {
        if (d < s) red[d] += red[d + s];
        __syncthreads();
    }
    if (d == 0) {
        float prob = sigf(red[0] + cls_b[0]);
        float op = sub_p[r] * prob;               // valid already folded into sub_p
        objp_out[r] = op;
        atomicAdd(&lastE_out[(size_t)b * NE + obj_idx[r]], op);
    }
    trans_out[(size_t)r * DD + d] = tr;
}

// ---------------- sparse hist: h_bk[b,k] = sum_m [obj0==topk1[b,k]] trans0*objp0 ----------------
__global__ void hist_gather_kernel(const int* __restrict__ topk, const int* __restrict__ obj_idx0,
                                   const float* __restrict__ trans0, const float* __restrict__ objp0,
                                   float* __restrict__ hbk) {
    int bk = blockIdx.x;           // 24 blocks
    int d = threadIdx.x;           // 128 threads
    int b = bk / KK;
    int e = topk[bk];
    float acc = 0.0f;
    for (int m = 0; m < MM; ++m) {
        int r = b * MM + m;
        if (obj_idx0[r] == e) acc += trans0[(size_t)r * DD + d] * objp0[r];
    }
    hbk[bk * DD + d] = acc;
}

// ---------------- final normalization: v / max(v, 1) ----------------
__global__ void finalize_kernel(float* p, int n) {
    int i = blockIdx.x * blockDim.x + threadIdx.x;
    if (i < n) { float v = p[i]; p[i] = v > 1.0f ? 1.0f : v; }
}

// ---------------- host ----------------
static inline size_t alignup(size_t x) { return (x + 255) & ~(size_t)255; }

extern "C" void kernel_launch(void* const* d_in, const int* in_sizes, int n_in,
                              void* d_out, int out_size, void* d_ws, size_t ws_size,
                              hipStream_t stream) {
    const float* start    = (const float*)d_in[0];   // [8, 40000]
    const float* rel_emb  = (const float*)d_in[1];   // [512, 128]
    const float* step_W   = (const float*)d_in[2];   // [2, 128, 128]
    const float* step_b   = (const float*)d_in[3];   // [2, 128]
    const float* gru_W_ih = (const float*)d_in[4];   // [384, 128]
    const float* gru_W_hh = (const float*)d_in[5];   // [384, 128]
    const float* gru_b_ih = (const float*)d_in[6];   // [384]
    const float* gru_b_hh = (const float*)d_in[7];   // [384]
    const float* cls_w    = (const float*)d_in[8];   // [128]
    const float* cls_b    = (const float*)d_in[9];   // [1]
    const int*   query    = (const int*)d_in[10];    // [8]
    const int*   kb_trip  = (const int*)d_in[11];    // [T, 3]
    const int*   kb_range = (const int*)d_in[12];    // [40000, 2]
    float* out = (float*)d_out;                      // [8, 40000]

    // workspace carve (256B aligned chunks)
    char* w = (char*)d_ws;
    size_t off = 0;
    auto carve = [&](size_t bytes) { char* p = w + off; off += alignup(bytes); return p; };
    float* cq      = (float*)carve(2 * BSZ * DD * sizeof(float));
    int*   topk0   = (int*)  carve(BSZ * KK * sizeof(int));
    int*   topk1   = (int*)  carve(BSZ * KK * sizeof(int));
    float* X       = (float*)carve((size_t)NROWS * DD * sizeof(float));
    float* H       = (float*)carve((size_t)NROWS * DD * sizeof(float));
    float* GI      = (float*)carve((size_t)NROWS * G3D * sizeof(float));
    float* GH      = (float*)carve((size_t)NROWS * G3D * sizeof(float));
    float* trans0  = (float*)carve((size_t)NROWS * DD * sizeof(float));
    float* objp0   = (float*)carve(NROWS * sizeof(float));
    float* subp    = (float*)carve(NROWS * sizeof(float));
    int*   obj_idx = (int*)  carve(NROWS * sizeof(int));
    float* hbk     = (float*)carve(BSZ * KK * DD * sizeof(float));
    float* lastE1  = (float*)carve((size_t)BSZ * NE * sizeof(float));
    if (off > ws_size) return;   // insufficient scratch (fixed sizes -> deterministic)

    const int nE = BSZ * NE;                 // 320000
    const int nGemmWaves = 2 * (NROWS / 16) * (G3D / 16);   // 4608
    const int gemmBlocks = nGemmWaves / 4;                  // 128 threads = 4 waves/block

    // zero accumulators
    zero_kernel<<<(nE + 255) / 256, 256, 0, stream>>>(lastE1, nE);
    zero_kernel<<<(nE + 255) / 256, 256, 0, stream>>>(out, nE);

    // query transform for both steps
    cq_kernel<<<16, DD, 0, stream>>>(rel_emb, query, step_W, step_b, cq);

    // ---- step 0 ----
    topk3_kernel<<<BSZ, 256, 0, stream>>>(start, topk0);
    gather_kernel<<<NROWS, DD, 0, stream>>>(topk0, /*nvalid=*/1, start, rel_emb,
                                            kb_trip, kb_range, /*hbk=*/nullptr,
                                            X, H, subp, obj_idx);
    wmma_gemm_kernel<<<gemmBlocks, 128, 0, stream>>>(X, H, gru_W_ih, gru_W_hh, GI, GH);
    gru_gate_kernel<<<NROWS, DD, 0, stream>>>(GI, GH, H, gru_b_ih, gru_b_hh,
                                              cq /* step 0 */, cls_w, cls_b,
                                              subp, obj_idx, lastE1, trans0, objp0);

    // ---- step 1 ----
    topk3_kernel<<<BSZ, 256, 0, stream>>>(lastE1, topk1);
    hist_gather_kernel<<<BSZ * KK, DD, 0, stream>>>(topk1, obj_idx, trans0, objp0, hbk);
    gather_kernel<<<NROWS, DD, 0, stream>>>(topk1, /*nvalid=*/KK, lastE1, rel_emb,
                                            kb_trip, kb_range, hbk,
                                            X, H, subp, obj_idx);
    wmma_gemm_kernel<<<gemmBlocks, 128, 0, stream>>>(X, H, gru_W_ih, gru_W_hh, GI, GH);
    gru_gate_kernel<<<NROWS, DD, 0, stream>>>(GI, GH, H, gru_b_ih, gru_b_hh,
                                              cq + BSZ * DD /* step 1 */, cls_w, cls_b,
                                              subp, obj_idx, out, trans0, objp0);

    // last_e / max(last_e, 1)
    finalize_kernel<<<(nE + 255) / 256, 256, 0, stream>>>(out, nE);
}